// tAntiSymmetricLayer_83150566850572
// MI455X (gfx1250) — compile-verified
//
#include <hip/hip_runtime.h>

// tAntiSymmetricLayer for MI455X (gfx1250), fused single-pass fp32 WMMA.
// y = mode3( Ahat facewise@ mode3(x, M), Minv ) + bias,  Minv = M^T (M orthogonal).
// Traffic-optimal: read x once (512MB), write y once (512MB) -> ~46us @ 23.3TB/s.

typedef __attribute__((ext_vector_type(2))) float v2f;
typedef __attribute__((ext_vector_type(8))) float v8f;

#define GAMMA 1e-4f
#define NN 64      // tensor rows/cols
#define PP 65536   // batch columns
#define DD 32      // mode-3 depth

// D = A(16x4 f32) * B(4x16 f32) + C(16x16 f32)
__device__ __forceinline__ v8f wmma4(v2f a, v2f b, v8f c) {
  // (neg_a, A, neg_b, B, c_mod, C, reuse_a, reuse_b)
  return __builtin_amdgcn_wmma_f32_16x16x4_f32(false, a, false, b, (short)0, c,
                                               false, false);
}

// Ahat[k][i][n] = sum_m (w[i,n,m] + w[n,i,m]) * M[k,m]  - gamma*(i==n)
__global__ void prep_ahat_kernel(const float* __restrict__ w,
                                 const float* __restrict__ Mm,
                                 float* __restrict__ Ahat) {
  int idx = blockIdx.x * blockDim.x + threadIdx.x;
  if (idx >= NN * NN * DD) return;
  int n = idx & 63, i = (idx >> 6) & 63, k = idx >> 12;
  float s = 0.f;
#pragma unroll
  for (int m = 0; m < DD; ++m)
    s += (w[(i * NN + n) * DD + m] + w[(n * NN + i) * DD + m]) * Mm[k * DD + m];
  if (i == n) s -= GAMMA;
  Ahat[k * NN * NN + i * NN + n] = s;
}

// One workgroup = 8 waves = one 16-wide P tile, all 3 stages fused via LDS.
// LDS buffer xh: [entity(64)][p(16)][k(32)] floats = 128 KB (dynamic).
__global__ __launch_bounds__(256) void tlayer_kernel(
    const float* __restrict__ x, const float* __restrict__ bias,
    const float* __restrict__ Mm, const float* __restrict__ Ahat,
    float* __restrict__ y) {
  extern __shared__ float xh[];
  const int tid = threadIdx.x;
  const int wave = tid >> 5;
  const int lane = tid & 31;
  const int lh = lane & 15;  // lane within half-wave
  const int hi = lane >> 4;  // which half-wave (selects K-pair / M+8 rows)
  const int p0 = blockIdx.x * 16;

  // ---------------- stage A: xhat = x ×3 M (wave owns 8 values of n) -------
  // A-frag: rows p (M-dim), K = m pair; B-frag: B[m][khat] = M[khat][m]
  for (int nn = 0; nn < 8; ++nn) {
    int n = wave * 8 + nn;
    const float* xrow = x + ((size_t)n * PP + p0 + lh) * DD + 2 * hi;
    v8f c0 = {}, c1 = {};
#pragma unroll
    for (int m0 = 0; m0 < DD; m0 += 4) {
      v2f a = *(const v2f*)(xrow + m0);
      v2f b0 = *(const v2f*)(Mm + (0 * 16 + lh) * DD + m0 + 2 * hi);
      v2f b1 = *(const v2f*)(Mm + (1 * 16 + lh) * DD + m0 + 2 * hi);
      c0 = wmma4(a, b0, c0);
      c1 = wmma4(a, b1, c1);
    }
#pragma unroll
    for (int r = 0; r < 8; ++r) {
      int p = r + 8 * hi;
      xh[(n * 16 + p) * DD + lh] = c0[r];
      xh[(n * 16 + p) * DD + 16 + lh] = c1[r];
    }
  }
  __syncthreads();

  // ------- stage B: yhat_k^T = xhat_k^T @ Ahat_k^T, in place per slice -----
  // Wave owns 4 slices k; slice-k LDS is wave-private -> safe in-place.
  for (int kk = 0; kk < 4; ++kk) {
    int k = wave * 4 + kk;
    const float* Ak = Ahat + k * NN * NN;
    v8f d0 = {}, d1 = {}, d2 = {}, d3 = {};
#pragma unroll
    for (int n0 = 0; n0 < NN; n0 += 4) {
      int na = n0 + 2 * hi;
      v2f a;
      a.x = xh[(na * 16 + lh) * DD + k];
      a.y = xh[((na + 1) * 16 + lh) * DD + k];
      // B[n][i] = Ahat_k[i][n] : contiguous pair along n
      v2f b0 = *(const v2f*)(Ak + (0 * 16 + lh) * NN + na);
      v2f b1 = *(const v2f*)(Ak + (1 * 16 + lh) * NN + na);
      v2f b2 = *(const v2f*)(Ak + (2 * 16 + lh) * NN + na);
      v2f b3 = *(const v2f*)(Ak + (3 * 16 + lh) * NN + na);
      d0 = wmma4(a, b0, d0);
      d1 = wmma4(a, b1, d1);
      d2 = wmma4(a, b2, d2);
      d3 = wmma4(a, b3, d3);
    }
#pragma unroll
    for (int r = 0; r < 8; ++r) {
      int p = r + 8 * hi;
      xh[(((0 * 16 + lh) * 16) + p) * DD + k] = d0[r];
      xh[(((1 * 16 + lh) * 16) + p) * DD + k] = d1[r];
      xh[(((2 * 16 + lh) * 16) + p) * DD + k] = d2[r];
      xh[(((3 * 16 + lh) * 16) + p) * DD + k] = d3[r];
    }
  }
  __syncthreads();

  // ------- stage C: y = yhat ×3 Minv + bias, Minv[l][k] = M[k][l] ----------
  // Wave owns 8 values of i.  B[k][l] = M[k*32 + l] (M used row-major as-is).
  for (int ii = 0; ii < 8; ++ii) {
    int i = wave * 8 + ii;
    v8f e0 = {}, e1 = {};
#pragma unroll
    for (int k0 = 0; k0 < DD; k0 += 4) {
      int ka = k0 + 2 * hi;
      v2f a = *(const v2f*)(&xh[(i * 16 + lh) * DD + ka]);  // contiguous in k
      v2f b0, b1;
      b0.x = Mm[ka * DD + lh];
      b0.y = Mm[(ka + 1) * DD + lh];
      b1.x = Mm[ka * DD + 16 + lh];
      b1.y = Mm[(ka + 1) * DD + 16 + lh];
      e0 = wmma4(a, b0, e0);
      e1 = wmma4(a, b1, e1);
    }
    float bv0 = bias[i * DD + lh];
    float bv1 = bias[i * DD + 16 + lh];
#pragma unroll
    for (int r = 0; r < 8; ++r) {
      size_t p = (size_t)(p0 + r + 8 * hi);
      y[((size_t)i * PP + p) * DD + lh] = e0[r] + bv0;
      y[((size_t)i * PP + p) * DD + 16 + lh] = e1[r] + bv1;
    }
  }
}

extern "C" void kernel_launch(void* const* d_in, const int* in_sizes, int n_in,
                              void* d_out, int out_size, void* d_ws,
                              size_t ws_size, hipStream_t stream) {
  const float* x = (const float*)d_in[0];     // (64, 65536, 32)
  const float* w = (const float*)d_in[1];     // (64, 64, 32)
  const float* bias = (const float*)d_in[2];  // (64, 1, 32)
  const float* Mm = (const float*)d_in[3];    // (32, 32)
  float* y = (float*)d_out;                   // (64, 65536, 32)
  float* Ahat = (float*)d_ws;                 // 64*64*32 floats = 512 KB

  prep_ahat_kernel<<<(NN * NN * DD + 255) / 256, 256, 0, stream>>>(w, Mm, Ahat);
  tlayer_kernel<<<PP / 16, 256, NN * 16 * DD * sizeof(float), stream>>>(
      x, bias, Mm, Ahat, y);
}